// FeatureTransformer_19189913878891
// MI455X (gfx1250) — compile-verified
//
#include <hip/hip_runtime.h>

// ---------------- problem constants (match reference) ----------------
// B=8192, K=32, NUM_FEATURES=40960, D=1024
#define K_SLOTS    32
#define NUM_FEAT   40960
#define DIM        1024
#define THREADS    256          // 8 wave32 waves; each thread owns 4 output columns
#define GROUP_ROWS 8            // rows per async group (async fallback kernel)

// ---------------- CDNA5 feature availability -------------------------
#if __has_builtin(__builtin_amdgcn_tensor_load_to_lds) && \
    __has_builtin(__builtin_amdgcn_s_wait_tensorcnt)
#define USE_TDM 1
#else
#define USE_TDM 0
#endif
#if __has_builtin(__builtin_amdgcn_global_load_async_to_lds_b128) && \
    __has_builtin(__builtin_amdgcn_s_wait_asynccnt)
#define USE_ASYNC_LDS 1
#else
#define USE_ASYNC_LDS 0
#endif

// ---------------- types ----------------------------------------------
typedef int          v4i   __attribute__((vector_size(16)));
typedef unsigned int u32x4 __attribute__((vector_size(16)));
typedef int          i32x4 __attribute__((vector_size(16)));
typedef int          i32x8 __attribute__((vector_size(32)));
typedef __attribute__((address_space(1))) v4i*        as1_v4i_ptr;  // global
typedef __attribute__((address_space(3))) v4i*        as3_v4i_ptr;  // LDS
typedef __attribute__((address_space(3))) const void* as3_cvoid_ptr;

__device__ __forceinline__ unsigned lds_offset(const void* p) {
  // generic (shared) pointer -> LDS byte offset as used by DS/TDM addressing
  return (unsigned)(unsigned long long)(as3_cvoid_ptr)p;
}

// ---------------- TDM gather-mode descriptor issue -------------------
#if USE_TDM
// Gather 16 rows of W (row indices idx16[0..15], each row DIM fp32) into LDS
// at lds_byte_addr, rows packed consecutively (row r at +r*DIM*4 bytes).
// D# layout per CDNA5 ISA 8.3-8.7 (gather mode, 16-bit indices, data_size=4B).
__device__ __forceinline__ void tdm_gather16(const float* W, const int* idx16,
                                             unsigned lds_byte_addr) {
  const unsigned long long ga = (unsigned long long)(size_t)W;

  u32x4 g0;
  g0[0] = 0x80000001u;                                  // gather_mode=1 | idx16 | count=1
  g0[1] = lds_byte_addr;                                // lds_addr (bytes)
  g0[2] = (unsigned)(ga & 0xFFFFFFFFu);                 // global_addr[31:0]
  g0[3] = (unsigned)((ga >> 32) & 0x1FFFFFFu)           // global_addr[56:32]
          | (2u << 30);                                 // type=2 ("image")

  i32x8 g1;
  g1[0] = (int)(2u << 16);                              // data_size=4B; wg_mask=0
  g1[1] = (int)(((unsigned)DIM & 0xFFFFu) << 16);       // tensor_dim0[15:0]
  g1[2] = (int)((((unsigned)DIM >> 16) & 0xFFFFu)       // tensor_dim0[31:16]
          | (((unsigned)NUM_FEAT & 0xFFFFu) << 16));    // tensor_dim1[15:0]
  g1[3] = (int)((((unsigned)NUM_FEAT >> 16) & 0xFFFFu)  // tensor_dim1[31:16]
          | (((unsigned)DIM & 0xFFFFu) << 16));         // tile_dim0 = DIM
  g1[4] = 16;                                           // tile_dim1 = #valid indices
  g1[5] = DIM;                                          // tensor_dim0_stride[31:0]
  g1[6] = 0;                                            // stride[47:32] | dim1_stride (ignored)
  g1[7] = 0;

  i32x4 g2, g3;                                         // packed 16-bit row indices
  g2[0] = (idx16[0]  & 0xFFFF) | (idx16[1]  << 16);
  g2[1] = (idx16[2]  & 0xFFFF) | (idx16[3]  << 16);
  g2[2] = (idx16[4]  & 0xFFFF) | (idx16[5]  << 16);
  g2[3] = (idx16[6]  & 0xFFFF) | (idx16[7]  << 16);
  g3[0] = (idx16[8]  & 0xFFFF) | (idx16[9]  << 16);
  g3[1] = (idx16[10] & 0xFFFF) | (idx16[11] << 16);
  g3[2] = (idx16[12] & 0xFFFF) | (idx16[13] << 16);
  g3[3] = (idx16[14] & 0xFFFF) | (idx16[15] << 16);

  // 6-arg toolchain form: extra int32x8 slot (ISA VADDR4 is unused/NULL) -> zeros.
  const i32x8 gz = {0, 0, 0, 0, 0, 0, 0, 0};
  __builtin_amdgcn_tensor_load_to_lds(g0, g1, g2, g3, gz, /*cpol*/ 0);
}
#endif

// ---------------- TDM kernel (primary) -------------------------------
// One block per sample; wave 0 issues two TDM gather descriptors (16 rows
// each, 64KB per buffer); all 8 waves reduce from LDS with float4 FMAs.
__global__ __launch_bounds__(THREADS) void embag_tdm_kernel(
    const int*   __restrict__ fidx, const float* __restrict__ fval,
    const float* __restrict__ W,    const float* __restrict__ bias,
    float*       __restrict__ out) {
  const int b = blockIdx.x;
  const int t = threadIdx.x;

#if USE_TDM
  __shared__ float sbuf0[16 * DIM];   // 64 KB
  __shared__ float sbuf1[16 * DIM];   // 64 KB

  // Uniform scalar loads of the sample's 32 (idx, val) pairs (SGPR-resident).
  const long base = (long)b * K_SLOTS;
  int   idx[K_SLOTS];
  float val[K_SLOTS];
#pragma unroll
  for (int k = 0; k < K_SLOTS; ++k) {
    const int i = __builtin_amdgcn_readfirstlane(fidx[base + k]);
    const float v = fval[base + k];
    val[k] = (i < 0) ? 0.0f : v;                            // zero padded slots
    idx[k] = (i < 0) ? 0 : (i >= NUM_FEAT ? NUM_FEAT - 1 : i);
  }

  // Only wave 0 issues the DMA (scalar branch; TDM ignores EXEC once issued).
  if (__builtin_amdgcn_readfirstlane(t) < 32) {
    tdm_gather16(W, idx + 0,  lds_offset(sbuf0));
    tdm_gather16(W, idx + 16, lds_offset(sbuf1));
  }

  const int c = t * 4;
  float4 acc = *(const float4*)(bias + c);

  __builtin_amdgcn_s_wait_tensorcnt(1);   // wave0: first descriptor landed; others: no-op
  __syncthreads();                        // publish sbuf0 to all waves
#pragma unroll
  for (int r = 0; r < 16; ++r) {
    const float  v = val[r];
    const float4 w = *(const float4*)(sbuf0 + r * DIM + c);
    acc.x += v * w.x; acc.y += v * w.y; acc.z += v * w.z; acc.w += v * w.w;
  }

  __builtin_amdgcn_s_wait_tensorcnt(0);   // second descriptor landed
  __syncthreads();
#pragma unroll
  for (int r = 0; r < 16; ++r) {
    const float  v = val[16 + r];
    const float4 w = *(const float4*)(sbuf1 + r * DIM + c);
    acc.x += v * w.x; acc.y += v * w.y; acc.z += v * w.z; acc.w += v * w.w;
  }

  *(float4*)(out + (size_t)b * DIM + c) = acc;
#else
  // Fallback: direct float4 gathers from global (L2-resident weight table).
  __shared__ int   sidx[K_SLOTS];
  __shared__ float sval[K_SLOTS];
  if (t < K_SLOTS) {
    int   i = fidx[b * K_SLOTS + t];
    float v = fval[b * K_SLOTS + t];
    sval[t] = (i < 0) ? 0.0f : v;
    sidx[t] = (i < 0) ? 0 : (i >= NUM_FEAT ? NUM_FEAT - 1 : i);
  }
  __syncthreads();
  const int c = t * 4;
  float4 acc = *(const float4*)(bias + c);
#pragma unroll 8
  for (int k = 0; k < K_SLOTS; ++k) {
    const float  v = sval[k];
    const float4 w = *(const float4*)(W + (size_t)sidx[k] * DIM + c);
    acc.x += v * w.x; acc.y += v * w.y; acc.z += v * w.z; acc.w += v * w.w;
  }
  *(float4*)(out + (size_t)b * DIM + c) = acc;
#endif
}

// ---------------- async-to-LDS kernel (kept as compiled reference) ----
#if USE_ASYNC_LDS
__device__ __forceinline__ void async_cp16(const float* gsrc, float* lds_dst) {
  __builtin_amdgcn_global_load_async_to_lds_b128(
      (as1_v4i_ptr)(gsrc), (as3_v4i_ptr)(lds_dst), 0, 0);
}

__device__ __forceinline__ void stage_group(const float* __restrict__ W,
                                            const int* sidx, int g,
                                            float* buf, int c) {
#pragma unroll
  for (int r = 0; r < GROUP_ROWS; ++r) {
    const size_t row = (size_t)sidx[g * GROUP_ROWS + r];
    async_cp16(W + row * (size_t)DIM + c, buf + r * DIM + c);
  }
}

__device__ __forceinline__ void accum_lds_group(const float* buf, const float* sval,
                                                int g, int c, float4& acc) {
#pragma unroll
  for (int r = 0; r < GROUP_ROWS; ++r) {
    const float  v = sval[g * GROUP_ROWS + r];
    const float4 w = *(const float4*)(buf + r * DIM + c);
    acc.x += v * w.x; acc.y += v * w.y; acc.z += v * w.z; acc.w += v * w.w;
  }
}

__global__ __launch_bounds__(THREADS) void embag_async_kernel(
    const int*   __restrict__ fidx, const float* __restrict__ fval,
    const float* __restrict__ W,    const float* __restrict__ bias,
    float*       __restrict__ out) {
  __shared__ float sbuf0[GROUP_ROWS * DIM];
  __shared__ float sbuf1[GROUP_ROWS * DIM];
  __shared__ int   sidx[K_SLOTS];
  __shared__ float sval[K_SLOTS];

  const int b = blockIdx.x;
  const int t = threadIdx.x;
  if (t < K_SLOTS) {
    int   i = fidx[b * K_SLOTS + t];
    float v = fval[b * K_SLOTS + t];
    sval[t] = (i < 0) ? 0.0f : v;
    sidx[t] = (i < 0) ? 0 : (i >= NUM_FEAT ? NUM_FEAT - 1 : i);
  }
  __syncthreads();

  const int c = t * 4;
  float4 acc = *(const float4*)(bias + c);

  stage_group(W, sidx, 0, sbuf0, c);
  stage_group(W, sidx, 1, sbuf1, c);
  __builtin_amdgcn_s_wait_asynccnt(8);
  __syncthreads();
  accum_lds_group(sbuf0, sval, 0, c, acc);
  __syncthreads();
  stage_group(W, sidx, 2, sbuf0, c);
  __builtin_amdgcn_s_wait_asynccnt(8);
  __syncthreads();
  accum_lds_group(sbuf1, sval, 1, c, acc);
  __syncthreads();
  stage_group(W, sidx, 3, sbuf1, c);
  __builtin_amdgcn_s_wait_asynccnt(8);
  __syncthreads();
  accum_lds_group(sbuf0, sval, 2, c, acc);
  __builtin_amdgcn_s_wait_asynccnt(0);
  __syncthreads();
  accum_lds_group(sbuf1, sval, 3, c, acc);

  *(float4*)(out + (size_t)b * DIM + c) = acc;
}
#endif // USE_ASYNC_LDS

// ---------------- launch ---------------------------------------------
extern "C" void kernel_launch(void* const* d_in, const int* in_sizes, int n_in,
                              void* d_out, int out_size, void* d_ws, size_t ws_size,
                              hipStream_t stream) {
  (void)n_in; (void)out_size; (void)d_ws; (void)ws_size;
  const int*   fidx = (const int*)d_in[0];
  const float* fval = (const float*)d_in[1];
  const float* W    = (const float*)d_in[2];
  const float* bias = (const float*)d_in[3];
  float*       out  = (float*)d_out;

  const int B = in_sizes[0] / K_SLOTS;   // 8192
  embag_tdm_kernel<<<dim3(B), dim3(THREADS), 0, stream>>>(fidx, fval, W, bias, out);
}